// NTM_49572512530477
// MI455X (gfx1250) — compile-verified
//
#include <hip/hip_runtime.h>
#include <math.h>

// NTM single step for MI455X (gfx1250, wave32).
// Heavy work = 3 streaming touches of the 134MB memory matrix (~17us HBM floor
// @23.3TB/s; pass2 should hit the 192MB L2). Content addressing is done with
// V_WMMA_F32_16X16X4_F32 fused with the per-row norm so memory is read once.

#define NROWS 131072
#define MCOLS 256
#define PASS2_BLOCKS 2048

typedef __attribute__((ext_vector_type(2))) float v2f;
typedef __attribute__((ext_vector_type(8))) float v8f;

// ---- workspace layout (float offsets) ----
enum : int {
  WS_GATES = 0,                     // 4096 raw LSTM gates
  WS_C     = 4096,                  // 1024 cell state
  WS_H     = 5120,                  // 1024 hidden state
  WS_KEY   = 6144,                  // 2*256 keys (tanh applied)
  WS_ERASE = 6656,                  // 256
  WS_ADD   = 6912,                  // 256
  WS_RAW   = 7168,                  // 12 raw head scalars [h*6 + {beta,g,s0,s1,s2,gamma}]
  WS_SCAL  = 7180,                  // 14: b0,b1,g0,g1,s00,s01,s02,s10,s11,s12,gm0,gm1,kn0,kn1
  WS_MAXP  = 7200,                  // 2*1024 block max partials
  WS_SUMP  = 9248,                  // 2*512 block exp-sum partials
  WS_POWP  = 10272,                 // 2*512 block pow-sum partials
  WS_RED6  = 11296,                 // m0,m1,Z0,Z1,S0,S1
  WS_READV = 11302,                 // 256 read vector
  WS_L     = 16384,                 // 2*N content logits
  WS_E     = 16384 + 2*NROWS,       // 2*N exp(l - max)
  WS_WP    = 16384 + 4*NROWS,       // 2*N sharpened weights (unnormalized)
  WS_RPART = 16384 + 6*NROWS,       // PASS2_BLOCKS*256 read partials
};

// ---- wave32 reduce helpers ----
__device__ __forceinline__ float warpSum(float v) {
#pragma unroll
  for (int o = 16; o > 0; o >>= 1) v += __shfl_down(v, o, 32);
  return v;
}
__device__ __forceinline__ float warpMax(float v) {
#pragma unroll
  for (int o = 16; o > 0; o >>= 1) v = fmaxf(v, __shfl_down(v, o, 32));
  return v;
}
__device__ __forceinline__ float sigmoidf(float x) { return 1.f / (1.f + expf(-x)); }
__device__ __forceinline__ float softplusf(float x) { return x > 20.f ? x : log1pf(expf(x)); }

// ---- 1) LSTM gates: 4096 dots of len 768+1024, one wave per output ----
__global__ void k_gates(const float* __restrict__ x, const float* __restrict__ prev_read,
                        const float* __restrict__ h_state,
                        const float* __restrict__ W_ih, const float* __restrict__ b_ih,
                        const float* __restrict__ W_hh, const float* __restrict__ b_hh,
                        float* __restrict__ ws) {
  int tid = threadIdx.x, lane = tid & 31;
  int j = blockIdx.x * 8 + (tid >> 5);               // 512 blocks * 8 waves = 4096
  const float* wi = W_ih + (size_t)j * 768;
  const float* wh = W_hh + (size_t)j * 1024;
  float acc = 0.f;
  for (int k = lane; k < 768; k += 32) {
    float v = (k < 512) ? x[k] : prev_read[k - 512];
    acc += v * wi[k];
  }
  for (int k = lane; k < 1024; k += 32) acc += h_state[k] * wh[k];
  acc = warpSum(acc);
  if (lane == 0) ws[WS_GATES + j] = acc + b_ih[j] + b_hh[j];
}

// ---- 2) c,h from gates (torch order i,f,g,o) ----
__global__ void k_ch(const float* __restrict__ c_state, float* __restrict__ ws) {
  int j = blockIdx.x * blockDim.x + threadIdx.x;     // 4*256 = 1024
  float ig = ws[WS_GATES + j];
  float fg = ws[WS_GATES + 1024 + j];
  float gg = ws[WS_GATES + 2048 + j];
  float og = ws[WS_GATES + 3072 + j];
  float c = sigmoidf(fg) * c_state[j] + sigmoidf(ig) * tanhf(gg);
  ws[WS_C + j] = c;
  ws[WS_H + j] = sigmoidf(og) * tanhf(c);
}

// ---- 3) head projections: 1036 dots of len 1024 against c, one wave each ----
__global__ void k_headparams(const float* __restrict__ Wk, const float* __restrict__ bk,
                             const float* __restrict__ We, const float* __restrict__ be,
                             const float* __restrict__ Wa, const float* __restrict__ ba,
                             const float* __restrict__ Wbeta, const float* __restrict__ bbeta,
                             const float* __restrict__ Wg, const float* __restrict__ bg,
                             const float* __restrict__ Wsh, const float* __restrict__ bsh,
                             const float* __restrict__ Wgamma, const float* __restrict__ bgamma,
                             float* __restrict__ ws) {
  int tid = threadIdx.x, lane = tid & 31;
  int w = blockIdx.x * 8 + (tid >> 5);
  if (w >= 1036) return;                             // uniform per wave
  const float* c = ws + WS_C;
  const float* base;
  float bias;
  int kind, outIdx;                                  // kind: 0=tanh 1=sigmoid 2=raw
  if (w < 512) {                                     // keys, both heads
    base = Wk + (size_t)w * 1024; bias = bk[w]; kind = 0; outIdx = WS_KEY + w;
  } else if (w < 768) {                              // erase (head 1 only)
    int m = w - 512;
    base = We + (size_t)(256 + m) * 1024; bias = be[256 + m]; kind = 1; outIdx = WS_ERASE + m;
  } else if (w < 1024) {                             // add (head 1 only)
    int m = w - 768;
    base = Wa + (size_t)(256 + m) * 1024; bias = ba[256 + m]; kind = 0; outIdx = WS_ADD + m;
  } else {                                           // 12 scalar projections
    int r = w - 1024, hh = r / 6, t = r % 6;
    kind = 2; outIdx = WS_RAW + r;
    if (t == 0)      { base = Wbeta  + (size_t)hh * 1024; bias = bbeta[hh]; }
    else if (t == 1) { base = Wg     + (size_t)hh * 1024; bias = bg[hh]; }
    else if (t <= 4) { base = Wsh + (size_t)(hh * 3 + (t - 2)) * 1024; bias = bsh[hh * 3 + t - 2]; }
    else             { base = Wgamma + (size_t)hh * 1024; bias = bgamma[hh]; }
  }
  float acc = 0.f;
  for (int k = lane; k < 1024; k += 32) acc += c[k] * base[k];
  acc = warpSum(acc);
  if (lane == 0) {
    float d = acc + bias;
    ws[outIdx] = (kind == 0) ? tanhf(d) : (kind == 1 ? sigmoidf(d) : d);
  }
}

// ---- 4) key norms + scalar activations ----
__global__ void k_finalize(float* __restrict__ ws) {
  int tid = threadIdx.x, lane = tid & 31, hw = tid >> 5;   // 64 threads = 2 waves
  if (hw < 2) {
    float s = 0.f;
    for (int k = lane; k < 256; k += 32) { float v = ws[WS_KEY + hw * 256 + k]; s += v * v; }
    s = warpSum(s);
    if (lane == 0) ws[WS_SCAL + 12 + hw] = fmaxf(sqrtf(s), 1e-8f);   // kn (EPS_COS clamp)
  }
  __syncthreads();
  if (tid < 2) {
    int h = tid;
    const float* r = ws + WS_RAW + h * 6;
    ws[WS_SCAL + h]     = softplusf(r[0]);                  // beta
    ws[WS_SCAL + 2 + h] = sigmoidf(r[1]);                   // gate
    float m = fmaxf(r[2], fmaxf(r[3], r[4]));
    float e0 = expf(r[2] - m), e1 = expf(r[3] - m), e2 = expf(r[4] - m);
    float inv = 1.f / (e0 + e1 + e2);
    ws[WS_SCAL + 4 + 3 * h + 0] = e0 * inv;                 // shift softmax
    ws[WS_SCAL + 4 + 3 * h + 1] = e1 * inv;
    ws[WS_SCAL + 4 + 3 * h + 2] = e2 * inv;
    ws[WS_SCAL + 10 + h] = 1.f + softplusf(r[5]);           // gamma
  }
}

// ---- 5) WMMA streaming pass: dots vs both keys + row norms -> logits + block max ----
// Per wave: 16-row tile, A = mem tile in 16x4 f32 striping (lane=row, K split
// across lane halves), B columns {0,1} = key0,key1. Row sum-of-squares is fused
// on the same A registers. B is loaded unconditionally from an aliased LDS slot
// and zeroed with a branchless select so the hot loop has no EXEC divergence.
__global__ void k_pass1(const float* __restrict__ mem, float* __restrict__ ws) {
  __shared__ float sKey[512];
  __shared__ float sD[8][16][16];
  __shared__ float sMax[8][2];
  int tid = threadIdx.x, lane = tid & 31, wv = tid >> 5;
  sKey[tid] = ws[WS_KEY + tid];
  sKey[tid + 256] = ws[WS_KEY + 256 + tid];
  __syncthreads();
  float beta0 = ws[WS_SCAL + 0], beta1 = ws[WS_SCAL + 1];
  float kn0 = ws[WS_SCAL + 12], kn1 = ws[WS_SCAL + 13];
  int rowBase = (blockIdx.x * 8 + wv) * 16;                 // 1024 blocks * 8 * 16 = N
  int r16 = lane & 15;
  int khalf = (lane >> 4) * 2;                              // K offset 0 or 2
  const bool keyLane = (r16 < 2);                           // lanes feeding B cols 0,1
  const float* bsrc = &sKey[(r16 & 1) * 256 + khalf];       // valid alias for all lanes
  const float* rowp = mem + (size_t)(rowBase + r16) * MCOLS;
  v8f acc = (v8f)(0.f);
  float sq = 0.f;
#pragma unroll 4
  for (int k0 = 0; k0 < 256; k0 += 4) {
    v2f a = *(const v2f*)(rowp + k0 + khalf);               // A: (row, K=k0+khalf, +1)
    sq += a.x * a.x + a.y * a.y;
    v2f braw = *(const v2f*)(bsrc + k0);                    // unconditional LDS load
    v2f b;
    b.x = keyLane ? braw.x : 0.f;                           // v_cndmask, no EXEC games
    b.y = keyLane ? braw.y : 0.f;
    acc = __builtin_amdgcn_wmma_f32_16x16x4_f32(false, a, false, b, (short)0, acc,
                                                false, false);
  }
  sq += __shfl_xor(sq, 16, 32);                             // combine K halves per row
#pragma unroll
  for (int j = 0; j < 8; ++j)                               // D: lanes<16 M=j, lanes>=16 M=j+8
    sD[wv][(lane < 16) ? j : j + 8][r16] = acc[j];
  __syncthreads();
  float m0 = -INFINITY, m1 = -INFINITY;
  if (lane < 16) {
    int row = rowBase + lane;
    float dot0 = sD[wv][lane][0];
    float dot1 = sD[wv][lane][1];
    float mn = fmaxf(sqrtf(sq), 1e-8f);
    float l0 = beta0 * dot0 / (kn0 * mn);
    float l1 = beta1 * dot1 / (kn1 * mn);
    ws[WS_L + row] = l0;
    ws[WS_L + NROWS + row] = l1;
    m0 = l0; m1 = l1;
  }
  m0 = warpMax(m0); m1 = warpMax(m1);
  if (lane == 0) { sMax[wv][0] = m0; sMax[wv][1] = m1; }
  __syncthreads();
  if (tid == 0) {
    float a0 = -INFINITY, a1 = -INFINITY;
    for (int i = 0; i < 8; ++i) { a0 = fmaxf(a0, sMax[i][0]); a1 = fmaxf(a1, sMax[i][1]); }
    ws[WS_MAXP + blockIdx.x] = a0;
    ws[WS_MAXP + 1024 + blockIdx.x] = a1;
  }
}

// ---- 6) global max over 1024 partials (both heads) ----
__global__ void k_reduce_max(float* __restrict__ ws) {
  __shared__ float sbuf[8];
  int tid = threadIdx.x;
  for (int h = 0; h < 2; ++h) {
    float v = -INFINITY;
    for (int i = tid; i < 1024; i += 256) v = fmaxf(v, ws[WS_MAXP + h * 1024 + i]);
    v = warpMax(v);
    if ((tid & 31) == 0) sbuf[tid >> 5] = v;
    __syncthreads();
    if (tid == 0) {
      float a = sbuf[0];
      for (int i = 1; i < 8; ++i) a = fmaxf(a, sbuf[i]);
      ws[WS_RED6 + h] = a;
    }
    __syncthreads();
  }
}

// ---- 7) exp(l - max) + block partial sums ----
__global__ void k_exp(float* __restrict__ ws) {
  __shared__ float sbuf[8];
  int tid = threadIdx.x;
  int n = blockIdx.x * 256 + tid;                           // 512 blocks
  float m0 = ws[WS_RED6], m1 = ws[WS_RED6 + 1];
  float e0 = expf(ws[WS_L + n] - m0);
  float e1 = expf(ws[WS_L + NROWS + n] - m1);
  ws[WS_E + n] = e0;
  ws[WS_E + NROWS + n] = e1;
  float s = warpSum(e0);
  if ((tid & 31) == 0) sbuf[tid >> 5] = s;
  __syncthreads();
  if (tid == 0) { float a = 0; for (int i = 0; i < 8; ++i) a += sbuf[i]; ws[WS_SUMP + blockIdx.x] = a; }
  __syncthreads();
  s = warpSum(e1);
  if ((tid & 31) == 0) sbuf[tid >> 5] = s;
  __syncthreads();
  if (tid == 0) { float a = 0; for (int i = 0; i < 8; ++i) a += sbuf[i]; ws[WS_SUMP + 512 + blockIdx.x] = a; }
}

// ---- 8/10) sum 512 partials per head ----
__global__ void k_reduce_sum512(float* __restrict__ ws, int inOff, int outOff) {
  __shared__ float sbuf[8];
  int tid = threadIdx.x;
  for (int h = 0; h < 2; ++h) {
    float v = ws[inOff + h * 512 + tid] + ws[inOff + h * 512 + 256 + tid];
    v = warpSum(v);
    if ((tid & 31) == 0) sbuf[tid >> 5] = v;
    __syncthreads();
    if (tid == 0) { float a = 0; for (int i = 0; i < 8; ++i) a += sbuf[i]; ws[outOff + h] = a; }
    __syncthreads();
  }
}

// ---- 9) fused gate-interp + circular 3-tap shift + pow(gamma) + partial sums ----
__global__ void k_wp(const float* __restrict__ prev_w, float* __restrict__ ws) {
  __shared__ float sbuf[8];
  int tid = threadIdx.x;
  int n = blockIdx.x * 256 + tid;                           // 512 blocks
  int nm = (n - 1) & (NROWS - 1);
  int np = (n + 1) & (NROWS - 1);
  float sum0, sum1;
  {
    float g = ws[WS_SCAL + 2], invZ = 1.f / ws[WS_RED6 + 2], gm = ws[WS_SCAL + 10];
    float a = g * ws[WS_E + nm] * invZ + (1.f - g) * prev_w[nm];
    float b = g * ws[WS_E + n ] * invZ + (1.f - g) * prev_w[n];
    float c = g * ws[WS_E + np] * invZ + (1.f - g) * prev_w[np];
    float sw = ws[WS_SCAL + 4] * a + ws[WS_SCAL + 5] * b + ws[WS_SCAL + 6] * c;
    float wp = powf(fmaxf(sw, 0.f), gm);
    ws[WS_WP + n] = wp;
    sum0 = wp;
  }
  {
    float g = ws[WS_SCAL + 3], invZ = 1.f / ws[WS_RED6 + 3], gm = ws[WS_SCAL + 11];
    const float* pw = prev_w + NROWS;
    float a = g * ws[WS_E + NROWS + nm] * invZ + (1.f - g) * pw[nm];
    float b = g * ws[WS_E + NROWS + n ] * invZ + (1.f - g) * pw[n];
    float c = g * ws[WS_E + NROWS + np] * invZ + (1.f - g) * pw[np];
    float sw = ws[WS_SCAL + 7] * a + ws[WS_SCAL + 8] * b + ws[WS_SCAL + 9] * c;
    float wp = powf(fmaxf(sw, 0.f), gm);
    ws[WS_WP + NROWS + n] = wp;
    sum1 = wp;
  }
  float s = warpSum(sum0);
  if ((tid & 31) == 0) sbuf[tid >> 5] = s;
  __syncthreads();
  if (tid == 0) { float a = 0; for (int i = 0; i < 8; ++i) a += sbuf[i]; ws[WS_POWP + blockIdx.x] = a; }
  __syncthreads();
  s = warpSum(sum1);
  if ((tid & 31) == 0) sbuf[tid >> 5] = s;
  __syncthreads();
  if (tid == 0) { float a = 0; for (int i = 0; i < 8; ++i) a += sbuf[i]; ws[WS_POWP + 512 + blockIdx.x] = a; }
}

// ---- 11) streaming pass 2: read-vector partials + erase/add update (float4) ----
__global__ void k_pass2(const float* __restrict__ mem, float* __restrict__ out_mem,
                        float* __restrict__ ws) {
  __shared__ float sRed[256 * 4];
  int tid = threadIdx.x;
  int c4 = tid & 63;                                        // which float4 column group
  int rsub = tid >> 6;                                      // 0..3 row sub-stripes
  float invS0 = 1.f / (ws[WS_RED6 + 4] + 1e-16f);
  float invS1 = 1.f / (ws[WS_RED6 + 5] + 1e-16f);
  int col = c4 * 4;
  float4 er = *(const float4*)(ws + WS_ERASE + col);
  float4 ad = *(const float4*)(ws + WS_ADD + col);
  float4 racc = make_float4(0.f, 0.f, 0.f, 0.f);
  int base = blockIdx.x * 64;                               // 2048 blocks * 64 rows = N
  for (int i = 0; i < 16; ++i) {
    int n = base + rsub * 16 + i;
    const float* rowp = mem + (size_t)n * MCOLS + col;
    __builtin_prefetch(rowp + MCOLS, 0, 1);                 // global_prefetch next row
    float wr = ws[WS_WP + n] * invS0;
    float ww = ws[WS_WP + NROWS + n] * invS1;
    float4 v = *(const float4*)rowp;
    racc.x += wr * v.x; racc.y += wr * v.y; racc.z += wr * v.z; racc.w += wr * v.w;
    float4 o;
    o.x = v.x * (1.f - ww * er.x) + ww * ad.x;
    o.y = v.y * (1.f - ww * er.y) + ww * ad.y;
    o.z = v.z * (1.f - ww * er.z) + ww * ad.z;
    o.w = v.w * (1.f - ww * er.w) + ww * ad.w;
    *(float4*)(out_mem + (size_t)n * MCOLS + col) = o;
  }
  sRed[tid * 4 + 0] = racc.x; sRed[tid * 4 + 1] = racc.y;
  sRed[tid * 4 + 2] = racc.z; sRed[tid * 4 + 3] = racc.w;
  __syncthreads();
  if (tid < 64) {
#pragma unroll
    for (int k = 0; k < 4; ++k) {
      float s = 0.f;
      for (int r = 0; r < 4; ++r) s += sRed[(r * 64 + tid) * 4 + k];
      ws[WS_RPART + blockIdx.x * 256 + tid * 4 + k] = s;
    }
  }
}

// ---- 12) reduce read partials over 2048 blocks ----
__global__ void k_reduce_read(float* __restrict__ ws) {
  int t = threadIdx.x;                                      // 256 threads = 256 columns
  float s = 0.f;
  for (int b = 0; b < PASS2_BLOCKS; ++b) s += ws[WS_RPART + b * 256 + t];
  ws[WS_READV + t] = s;
}

// ---- 13) output layer: 512 dots of len 1280, one wave each ----
__global__ void k_out(const float* __restrict__ W_out, const float* __restrict__ b_out,
                      const float* __restrict__ ws, float* __restrict__ out) {
  int tid = threadIdx.x, lane = tid & 31;
  int j = blockIdx.x * 8 + (tid >> 5);                      // 64 blocks * 8 = 512
  const float* wr = W_out + (size_t)j * 1280;
  float acc = 0.f;
  for (int k = lane; k < 1280; k += 32) {
    float q = (k < 1024) ? ws[WS_H + k] : ws[WS_READV + k - 1024];
    acc += q * wr[k];
  }
  acc = warpSum(acc);
  if (lane == 0) out[j] = sigmoidf(acc + b_out[j]);
}

extern "C" void kernel_launch(void* const* d_in, const int* in_sizes, int n_in,
                              void* d_out, int out_size, void* d_ws, size_t ws_size,
                              hipStream_t stream) {
  const float* x         = (const float*)d_in[0];
  const float* prev_read = (const float*)d_in[1];
  const float* h_state   = (const float*)d_in[2];
  const float* c_state   = (const float*)d_in[3];
  const float* memory    = (const float*)d_in[4];
  const float* prev_w    = (const float*)d_in[5];
  const float* W_ih      = (const float*)d_in[6];
  const float* b_ih      = (const float*)d_in[7];
  const float* W_hh      = (const float*)d_in[8];
  const float* b_hh      = (const float*)d_in[9];
  const float* W_out     = (const float*)d_in[10];
  const float* b_out     = (const float*)d_in[11];
  const float* Wk        = (const float*)d_in[12];
  const float* bk        = (const float*)d_in[13];
  const float* Wbeta     = (const float*)d_in[14];
  const float* bbeta     = (const float*)d_in[15];
  const float* Wg        = (const float*)d_in[16];
  const float* bg        = (const float*)d_in[17];
  const float* Wsh       = (const float*)d_in[18];
  const float* bsh       = (const float*)d_in[19];
  const float* Wgamma    = (const float*)d_in[20];
  const float* bgamma    = (const float*)d_in[21];
  const float* We        = (const float*)d_in[22];
  const float* be        = (const float*)d_in[23];
  const float* Wa        = (const float*)d_in[24];
  const float* ba        = (const float*)d_in[25];
  float* out = (float*)d_out;
  float* ws  = (float*)d_ws;

  k_gates<<<512, 256, 0, stream>>>(x, prev_read, h_state, W_ih, b_ih, W_hh, b_hh, ws);
  k_ch<<<4, 256, 0, stream>>>(c_state, ws);
  k_headparams<<<130, 256, 0, stream>>>(Wk, bk, We, be, Wa, ba, Wbeta, bbeta,
                                        Wg, bg, Wsh, bsh, Wgamma, bgamma, ws);
  k_finalize<<<1, 64, 0, stream>>>(ws);
  k_pass1<<<1024, 256, 0, stream>>>(memory, ws);
  k_reduce_max<<<1, 256, 0, stream>>>(ws);
  k_exp<<<512, 256, 0, stream>>>(ws);
  k_reduce_sum512<<<1, 256, 0, stream>>>(ws, WS_SUMP, WS_RED6 + 2);
  k_wp<<<512, 256, 0, stream>>>(prev_w, ws);
  k_reduce_sum512<<<1, 256, 0, stream>>>(ws, WS_POWP, WS_RED6 + 4);
  k_pass2<<<PASS2_BLOCKS, 256, 0, stream>>>(memory, out + 512, ws);
  k_reduce_read<<<1, 256, 0, stream>>>(ws);
  k_out<<<64, 256, 0, stream>>>(W_out, b_out, ws, out);
}